// GlobalConcatAttention_69483980915149
// MI455X (gfx1250) — compile-verified
//
#include <hip/hip_runtime.h>
#include <hip/hip_bf16.h>
#include <stdint.h>

// ---------------------------------------------------------------------------
// GlobalConcatAttention on MI455X (gfx1250, wave32, WMMA + TDM).
//
//   1) offsets/counts/pos from sorted batch array (no atomics)
//   2) position-major output permutation via 80x2048 rank scan
//   3) fused tail weight  Wf = w_out @ out_proj_w, bf = w_out@op_b + b_out
//   4) QKV GEMM [N,256]->[N,768]: 64-row A tile staged once per block
//      (x-half via TENSOR_LOAD_TO_LDS with pad->pitch 129, metal-half gathered
//      on the vector path in parallel), all 48 column tiles swept by 8 waves,
//      V_WMMA_F32_16X16X4_F32 inner product.
//   5) per-(graph,head) dense attention, scores and ctx via WMMA, LDS-resident
//   6) fused out GEMM [N,256]x[256,256] (A tile via TDM), scatter rows into
//      permuted d_out
// ---------------------------------------------------------------------------

#define BG   2048
#define EDIM 256
#define QKVD 768
#define LMAX 80

typedef __attribute__((ext_vector_type(2))) float v2f;
typedef __attribute__((ext_vector_type(8))) float v8f;
typedef __attribute__((ext_vector_type(4))) unsigned int u32x4;
typedef __attribute__((ext_vector_type(8))) int i32x8;
typedef __attribute__((ext_vector_type(4))) int i32x4;

__device__ __forceinline__ v8f wmma_f32_4(v2f a, v2f b, v8f c) {
  // D(16x16,f32) += A(16x4,f32) * B(4x16,f32)
  return __builtin_amdgcn_wmma_f32_16x16x4_f32(
      false, a, false, b, (short)0, c, false, false);
}

// TDM: load a [tile_rows x row_elems] f32 tile (row-major, stride row_elems)
// into LDS, padding 1 DWORD per row (LDS pitch = row_elems+1).
// pad_code: 6 -> pad every 128 DWORDs, 7 -> every 256 DWORDs.
__device__ __forceinline__ void tdm_load_2d(unsigned int lds_byte_addr,
                                            const void* gptr,
                                            unsigned int row_elems,
                                            unsigned int tensor_rows,
                                            unsigned int tile_rows,
                                            unsigned int pad_code) {
  unsigned long long ga = (unsigned long long)gptr;
  u32x4 g0;
  g0[0] = 1u;                                    // count=1, user descriptor
  g0[1] = lds_byte_addr;                         // lds_addr [63:32]
  g0[2] = (unsigned int)ga;                      // global_addr lo
  g0[3] = (unsigned int)((ga >> 32) & 0x01FFFFFFu) | (2u << 30);  // addr hi | type=2
  i32x8 g1;
  // data_size=4B (2), pad_enable, pad_interval=pad_code, pad_amount=0 (1 DWORD)
  g1[0] = (int)((2u << 16) | (1u << 20) | (pad_code << 22));
  g1[1] = (int)(row_elems << 16);                // tensor_dim0 low16 @ [63:48]
  g1[2] = (int)((tensor_rows & 0xFFFFu) << 16);  // dim0 hi16=0 | tensor_dim1 lo16
  g1[3] = (int)(((tensor_rows >> 16) & 0xFFFFu) | (row_elems << 16)); // dim1 hi | tile_dim0
  g1[4] = (int)tile_rows;                        // tile_dim1 | tile_dim2=0
  g1[5] = (int)row_elems;                        // tensor_dim0_stride low32
  g1[6] = 0;                                     // stride hi | dim1_stride lo
  g1[7] = 0;
  i32x4 z4 = {0, 0, 0, 0};
  i32x8 z8 = {0, 0, 0, 0, 0, 0, 0, 0};
  // 6-arg toolchain form: (g0, g1, g2, g3, g4, cpol)
  __builtin_amdgcn_tensor_load_to_lds(g0, g1, z4, z4, z8, 0);
}

// ---- setup: graph offsets / counts / node positions -----------------------
__global__ void k_offsets(const int* __restrict__ batch, int* __restrict__ offsets, int N) {
  int n = blockIdx.x * blockDim.x + threadIdx.x;
  if (n >= N) return;
  if (n == 0 || batch[n] != batch[n - 1]) offsets[batch[n]] = n;
}

__global__ void k_counts_pos(const int* __restrict__ batch, const int* __restrict__ offsets,
                             int* __restrict__ counts, int* __restrict__ pos, int N) {
  int t = blockIdx.x * blockDim.x + threadIdx.x;
  if (t < BG) counts[t] = ((t == BG - 1) ? N : offsets[t + 1]) - offsets[t];
  if (t < N)  pos[t] = t - offsets[batch[t]];
}

// output row of node (p,b) = rowstart[p] + #{b' < b : counts[b'] > p}
__global__ void k_rank(const int* __restrict__ counts, int* __restrict__ rank,
                       int* __restrict__ rowstart) {
  int p = threadIdx.x;                 // single block, 128 threads
  if (p < LMAX) {
    int acc = 0;
    for (int b = 0; b < BG; ++b) {
      rank[p * BG + b] = acc;
      acc += (counts[b] > p) ? 1 : 0;
    }
    rowstart[p] = acc;
  }
  __syncthreads();
  if (p == 0) {
    int run = 0;
    for (int q = 0; q < LMAX; ++q) { int c = rowstart[q]; rowstart[q] = run; run += c; }
  }
}

__global__ void k_outrow(const int* __restrict__ batch, const int* __restrict__ pos,
                         const int* __restrict__ rank, const int* __restrict__ rowstart,
                         int* __restrict__ outrow, int N) {
  int n = blockIdx.x * blockDim.x + threadIdx.x;
  if (n >= N) return;
  int p = pos[n], b = batch[n];
  outrow[n] = rowstart[p] + rank[p * BG + b];
}

// ---- fuse out_proj and w_out into one 256x256 weight ----------------------
__global__ void k_fusew(const float* __restrict__ w_out, const float* __restrict__ op_w,
                        const float* __restrict__ op_b, const float* __restrict__ b_out,
                        float* __restrict__ Wf, float* __restrict__ bf) {
  int t = blockIdx.x * blockDim.x + threadIdx.x;
  if (t >= EDIM * EDIM) return;
  int i = t >> 8, j = t & 255;
  float acc = 0.f;
  for (int k = 0; k < EDIM; ++k) acc += w_out[i * EDIM + k] * op_w[k * EDIM + j];
  Wf[t] = acc;
  if (j == 0) {
    float bb = 0.f;
    for (int k = 0; k < EDIM; ++k) bb += w_out[i * EDIM + k] * op_b[k];
    bf[i] = bb + b_out[i];
  }
}

// ---- QKV projection: Y[N,768] = concat(metal_x[g], x) @ W^T + b -----------
// 64-row A tile loaded ONCE per block (no column-split re-gather); 8 waves
// sweep 4 row-tiles x 48 col-tiles.
__global__ __launch_bounds__(256) void k_qkv(const float* __restrict__ x,
                                             const float* __restrict__ mx,
                                             const int* __restrict__ batch,
                                             const float* __restrict__ W,
                                             const float* __restrict__ bias,
                                             float* __restrict__ qkv, int N) {
  __shared__ float Am[64][129];        // metal half (gathered)
  __shared__ float Ax[64][129];        // x half (TDM, pad -> pitch 129)
  const int m0 = blockIdx.x * 64;
  const int tid = threadIdx.x;
  if (tid < 32) {                      // wave 0 issues the tensor DMA
    tdm_load_2d((unsigned int)(uintptr_t)&Ax[0][0],
                x + (size_t)m0 * 128, 128u, (unsigned int)N, 64u, 6u);
  }
  // overlap: irregular metal_x gather on the vector path
  for (int idx = tid; idx < 64 * 128; idx += 256) {
    int r = idx >> 7, c2 = idx & 127;
    int row = m0 + r;
    Am[r][c2] = (row < N) ? mx[(size_t)batch[row] * 128 + c2] : 0.f;
  }
  if (tid < 32) __builtin_amdgcn_s_wait_tensorcnt(0);
  __syncthreads();

  const int lane = tid & 31, wave = tid >> 5;
  const int rg = wave >> 2, cg = wave & 3;
  const int kl = (lane >> 4) << 1;     // lanes 0-15: K pair {0,1}; 16-31: {2,3}
  const int ml = lane & 15;
  const int mb = (lane < 16) ? 0 : 8;
  for (int rt = rg; rt < 4; rt += 2) {
    const int ar = rt * 16 + ml;
    for (int p = 0; p < 12; ++p) {
      const int n = p * 64 + cg * 16 + ml;
      const float* wrow = W + (size_t)n * EDIM;   // B[k][n] = W[n][k]
      v8f c = {};
      for (int k0 = 0; k0 < 128; k0 += 4) {
        int kb = k0 + kl;
        v2f a, b;
        a.x = Am[ar][kb]; a.y = Am[ar][kb + 1];
        b.x = wrow[kb];   b.y = wrow[kb + 1];
        c = wmma_f32_4(a, b, c);
      }
      for (int k0 = 0; k0 < 128; k0 += 4) {
        int kb = k0 + kl;
        v2f a, b;
        a.x = Ax[ar][kb];       a.y = Ax[ar][kb + 1];
        b.x = wrow[128 + kb];   b.y = wrow[128 + kb + 1];
        c = wmma_f32_4(a, b, c);
      }
      const float bn = bias[n];
      for (int j = 0; j < 8; ++j) {
        int row = m0 + rt * 16 + j + mb;
        if (row < N) qkv[(size_t)row * QKVD + n] = c[j] + bn;
      }
    }
  }
}

// ---- per-(graph, head) attention, fully LDS-resident, WMMA matmuls --------
__global__ __launch_bounds__(128) void k_attn(const float* __restrict__ qkv,
                                              const int* __restrict__ offsets,
                                              const int* __restrict__ counts,
                                              float* __restrict__ ctx) {
  __shared__ float qs[LMAX][65];
  __shared__ float ks[LMAX][65];
  __shared__ float vs[LMAX][65];
  __shared__ float ss[LMAX][81];       // scores -> softmax probs
  const int g = blockIdx.x, h = blockIdx.y;
  const int off = offsets[g];
  const int L = counts[g];
  const int tid = threadIdx.x, lane = tid & 31, wave = tid >> 5;

  // zero padding rows (prevents NaN from 0 * garbage inside WMMA dot)
  for (int idx = L * 64 + tid; idx < LMAX * 64; idx += 128) {
    int i = idx >> 6, d = idx & 63;
    qs[i][d] = 0.f; ks[i][d] = 0.f; vs[i][d] = 0.f;
  }
  for (int idx = tid; idx < L * 64; idx += 128) {
    int i = idx >> 6, d = idx & 63;
    const float* base = qkv + (size_t)(off + i) * QKVD + h * 64 + d;
    qs[i][d] = base[0];
    ks[i][d] = base[256];
    vs[i][d] = base[512];
  }
  __syncthreads();

  const int mt = (L + 15) >> 4;        // <= 5 tiles per side
  const int kl = (lane >> 4) << 1;
  // scores = Q @ K^T * 1/sqrt(64); wave-uniform tile loop (EXEC all-ones)
  for (int t = wave; t < mt * mt; t += 4) {
    int mi = t / mt, ni = t % mt;
    int am = mi * 16 + (lane & 15);
    int bn = ni * 16 + (lane & 15);
    v8f c = {};
    for (int k0 = 0; k0 < 64; k0 += 4) {
      int kb = k0 + kl;
      v2f a, b;
      a.x = qs[am][kb]; a.y = qs[am][kb + 1];
      b.x = ks[bn][kb]; b.y = ks[bn][kb + 1];
      c = wmma_f32_4(a, b, c);
    }
    int mo = mi * 16 + ((lane < 16) ? 0 : 8);
    for (int j = 0; j < 8; ++j) ss[mo + j][ni * 16 + (lane & 15)] = c[j] * 0.125f;
  }
  __syncthreads();

  // row softmax over the L valid keys; zero probability for padded keys
  for (int i = wave; i < L; i += 4) {
    float m = -3.4e38f;
    for (int j = lane; j < L; j += 32) m = fmaxf(m, ss[i][j]);
    for (int o = 16; o; o >>= 1) m = fmaxf(m, __shfl_xor(m, o, 32));
    float s = 0.f;
    for (int j = lane; j < L; j += 32) { float e = __expf(ss[i][j] - m); ss[i][j] = e; s += e; }
    for (int o = 16; o; o >>= 1) s += __shfl_xor(s, o, 32);
    float inv = 1.f / s;
    for (int j = lane; j < L; j += 32) ss[i][j] *= inv;
    for (int j = L + lane; j < LMAX; j += 32) ss[i][j] = 0.f;
  }
  __syncthreads();

  // ctx = P @ V  (K padded to mt*16; padded P cols and V rows are zero)
  const int Lp = mt * 16;
  for (int t = wave; t < mt * 4; t += 4) {
    int mi = t >> 2, ni = t & 3;
    int am = mi * 16 + (lane & 15);
    int bn = ni * 16 + (lane & 15);
    v8f c = {};
    for (int k0 = 0; k0 < Lp; k0 += 4) {
      int kb = k0 + kl;
      v2f a, b;
      a.x = ss[am][kb];  a.y = ss[am][kb + 1];
      b.x = vs[kb][bn];  b.y = vs[kb + 1][bn];
      c = wmma_f32_4(a, b, c);
    }
    int mo = mi * 16 + ((lane < 16) ? 0 : 8);
    for (int j = 0; j < 8; ++j) {
      int row = mo + j;
      if (row < L) ctx[(size_t)(off + row) * EDIM + h * 64 + bn] = c[j];
    }
  }
}

// ---- fused out GEMM: d_out[outrow[n]] = ctx[n] @ Wf^T + bf ----------------
__global__ __launch_bounds__(256) void k_out(const float* __restrict__ ctx,
                                             const float* __restrict__ Wf,
                                             const float* __restrict__ bf,
                                             const int* __restrict__ outrow,
                                             float* __restrict__ out, int N) {
  __shared__ float As[64][257];        // TDM pad -> pitch 257
  const int m0 = blockIdx.x * 64;
  const int tid = threadIdx.x;
  if (tid < 32) {
    tdm_load_2d((unsigned int)(uintptr_t)&As[0][0],
                ctx + (size_t)m0 * 256, 256u, (unsigned int)N, 64u, 7u);
    __builtin_amdgcn_s_wait_tensorcnt(0);
  }
  __syncthreads();

  const int lane = tid & 31, wave = tid >> 5;
  const int rg = wave >> 2, cg = wave & 3;
  const int kl = (lane >> 4) << 1;
  const int ml = lane & 15;
  const int mb = (lane < 16) ? 0 : 8;
  for (int rt = rg; rt < 4; rt += 2) {
    const int ar = rt * 16 + ml;
    for (int p = 0; p < 4; ++p) {
      const int n = p * 64 + cg * 16 + ml;
      const float* wrow = Wf + (size_t)n * EDIM;
      v8f c = {};
      for (int k0 = 0; k0 < EDIM; k0 += 4) {
        int kb = k0 + kl;
        v2f a, b;
        a.x = As[ar][kb]; a.y = As[ar][kb + 1];
        b.x = wrow[kb];   b.y = wrow[kb + 1];
        c = wmma_f32_4(a, b, c);
      }
      const float bn = bf[n];
      for (int j = 0; j < 8; ++j) {
        int row = m0 + rt * 16 + j + mb;
        if (row < N) out[(size_t)outrow[row] * EDIM + n] = c[j] + bn;
      }
    }
  }
}

// ---------------------------------------------------------------------------
extern "C" void kernel_launch(void* const* d_in, const int* in_sizes, int n_in,
                              void* d_out, int out_size, void* d_ws, size_t ws_size,
                              hipStream_t stream) {
  const float* x    = (const float*)d_in[0];
  const float* mx   = (const float*)d_in[1];
  const float* ipw  = (const float*)d_in[2];
  const float* ipb  = (const float*)d_in[3];
  const float* opw  = (const float*)d_in[4];
  const float* opb  = (const float*)d_in[5];
  const float* wout = (const float*)d_in[6];
  const float* bout = (const float*)d_in[7];
  const int*   batch = (const int*)d_in[8];
  float* out = (float*)d_out;
  const int N = in_sizes[0] / 128;

  // workspace carve-out (256B aligned)
  char* p = (char*)d_ws;
  auto carve = [&](size_t bytes) { char* r = p; p += (bytes + 255) & ~(size_t)255; return r; };
  float* qkv      = (float*)carve((size_t)N * QKVD * sizeof(float));
  float* ctx      = (float*)carve((size_t)N * EDIM * sizeof(float));
  float* Wf       = (float*)carve((size_t)EDIM * EDIM * sizeof(float));
  float* bf       = (float*)carve(EDIM * sizeof(float));
  int*   offsets  = (int*)carve(BG * sizeof(int));
  int*   counts   = (int*)carve(BG * sizeof(int));
  int*   posArr   = (int*)carve((size_t)N * sizeof(int));
  int*   rank     = (int*)carve((size_t)LMAX * BG * sizeof(int));
  int*   rowstart = (int*)carve(LMAX * sizeof(int));
  int*   outrow   = (int*)carve((size_t)N * sizeof(int));

  const int nb = (N + 255) / 256;
  k_offsets<<<nb, 256, 0, stream>>>(batch, offsets, N);
  k_counts_pos<<<nb, 256, 0, stream>>>(batch, offsets, counts, posArr, N);
  k_rank<<<1, 128, 0, stream>>>(counts, rank, rowstart);
  k_outrow<<<nb, 256, 0, stream>>>(batch, posArr, rank, rowstart, outrow, N);
  k_fusew<<<256, 256, 0, stream>>>(wout, opw, opb, bout, Wf, bf);

  const int nb64 = (N + 63) / 64;
  k_qkv<<<nb64, 256, 0, stream>>>(x, mx, batch, ipw, ipb, qkv, N);

  dim3 ga(BG, 4);
  k_attn<<<ga, 128, 0, stream>>>(qkv, offsets, counts, ctx);

  k_out<<<nb64, 256, 0, stream>>>(ctx, Wf, bf, outrow, out, N);
}